// ABS_MAE_48198122995800
// MI455X (gfx1250) — compile-verified
//
#include <hip/hip_runtime.h>
#include <hip/hip_bf16.h>

typedef __attribute__((ext_vector_type(2))) float v2f;
typedef __attribute__((ext_vector_type(8))) float v8f;

#define C_CLASSES 1000
#define N_ROWS    128
#define BLOCK     256
#define NWAVES    (BLOCK / 32)

__global__ __launch_bounds__(BLOCK)
void row_loss_kernel(const float* __restrict__ outputs,
                     const float* __restrict__ conf,
                     const int*   __restrict__ index,
                     float*       __restrict__ row_res)
{
    const int row  = blockIdx.x;
    const int t    = threadIdx.x;
    const int lane = t & 31;
    const int wid  = t >> 5;

    const float* __restrict__ x = outputs + (size_t)row * C_CLASSES;
    const long long gidx = (long long)index[row];
    const float* __restrict__ c = conf + (size_t)gidx * C_CLASSES;

    // Prefetch the gathered confidence row (4 KB) while the max pass runs:
    // 256 threads x 16B stride touches every 128B line of the row.
    __builtin_prefetch(c + (t << 2), 0, 1);

    const int j0 = t, j1 = t + 256, j2 = t + 512, j3 = t + 768;
    const bool v3 = (j3 < C_CLASSES);

    // Keep the row in registers: one global read of x per element.
    const float x0 = x[j0];
    const float x1 = x[j1];
    const float x2 = x[j2];
    const float x3 = v3 ? x[j3] : -3.402823466e38f;

    // ---- pass 1: row max (wave shuffle tree + cross-wave via LDS) ----
    float lmax = fmaxf(fmaxf(x0, x1), fmaxf(x2, x3));
#pragma unroll
    for (int off = 16; off > 0; off >>= 1)
        lmax = fmaxf(lmax, __shfl_xor(lmax, off, 32));

    __shared__ float smax[NWAVES];
    __shared__ float parts[24];          // [quantity 0..2][wave 0..7]
    if (lane == 0) smax[wid] = lmax;
    __syncthreads();
    float rowmax = smax[0];
#pragma unroll
    for (int w = 1; w < NWAVES; ++w) rowmax = fmaxf(rowmax, smax[w]);

    // ---- pass 2: exp + gathered-confidence dot, all in registers ----
    const float e0 = __expf(x0 - rowmax);
    const float e1 = __expf(x1 - rowmax);
    const float e2 = __expf(x2 - rowmax);
    const float e3 = v3 ? __expf(x3 - rowmax) : 0.0f;

    const float c0 = c[j0];
    const float c1 = c[j1];
    const float c2 = c[j2];
    const float c3 = v3 ? c[j3] : 0.0f;

    float se = (e0 + e1) + (e2 + e3);             // softmax denominator partial
    float s1 = (c0 + c1) + (c2 + c3);             // sum conf partial
    float s2 = (e0 * c0 + e1 * c1) + (e2 * c2 + e3 * c3);  // dot partial

#pragma unroll
    for (int off = 16; off > 0; off >>= 1) {
        se += __shfl_xor(se, off, 32);
        s1 += __shfl_xor(s1, off, 32);
        s2 += __shfl_xor(s2, off, 32);
    }
    if (lane == 0) {
        parts[0 * 8 + wid] = se;
        parts[1 * 8 + wid] = s1;
        parts[2 * 8 + wid] = s2;
    }
    __syncthreads();

    // ---- block-level reduce of 3x8 partials with ONE v_wmma_f32_16x16x4_f32 ----
    // A(16x4): row m = partial-group, K from the lane layout
    //   (lanes 0-15 -> K=0,1 in VGPR0/1 ; lanes 16-31 -> K=2,3).
    // Row 2q   holds quantity q, waves 0..3
    // Row 2q+1 holds quantity q, waves 4..7
    // B = ones(4x16)  =>  D[m][*] = rowsum(m).
    // D layout: lane 0 holds D[M=v][N=0] in d[v], so thread 0 gets everything.
    const int m    = lane & 15;
    const int half = lane >> 4;
    const int q    = m >> 1;
    const int w0   = ((m & 1) << 2) + (half << 1);
    float ax = 0.0f, ay = 0.0f;
    if (q < 3) {
        ax = parts[q * 8 + w0];
        ay = parts[q * 8 + w0 + 1];
    }
    v2f a; a.x = ax;   a.y = ay;
    v2f b; b.x = 1.0f; b.y = 1.0f;
    v8f acc = {};
    v8f d = __builtin_amdgcn_wmma_f32_16x16x4_f32(
        /*neg_a=*/false, a, /*neg_b=*/false, b,
        /*c_mod=*/(short)0, acc, /*reuse_a=*/false, /*reuse_b=*/false);

    if (t == 0) {
        const float SE = d[0] + d[1];
        const float S1 = d[2] + d[3];
        const float S2 = d[4] + d[5];
        // loss row value: sum_j (2 - 2 p_j) * conf_j = 2*S1 - 2*S2/SE
        row_res[row] = 2.0f * S1 - 2.0f * S2 / SE;
    }
}

__global__ __launch_bounds__(N_ROWS)
void finalize_kernel(const float* __restrict__ row_res, float* __restrict__ out)
{
    __shared__ float s[N_ROWS];
    const int t = threadIdx.x;
    s[t] = row_res[t];
    __syncthreads();
#pragma unroll
    for (int off = N_ROWS / 2; off > 0; off >>= 1) {
        if (t < off) s[t] += s[t + off];
        __syncthreads();
    }
    if (t == 0) out[0] = s[0] * (1.0f / (float)N_ROWS);
}

extern "C" void kernel_launch(void* const* d_in, const int* in_sizes, int n_in,
                              void* d_out, int out_size, void* d_ws, size_t ws_size,
                              hipStream_t stream)
{
    const float* outputs = (const float*)d_in[0];  // [128, 1000] f32
    const float* conf    = (const float*)d_in[1];  // [50000, 1000] f32
    const int*   index   = (const int*)d_in[2];    // [128] i32
    float* row_res = (float*)d_ws;                 // 128 floats scratch

    row_loss_kernel<<<N_ROWS, BLOCK, 0, stream>>>(outputs, conf, index, row_res);
    finalize_kernel<<<1, N_ROWS, 0, stream>>>(row_res, (float*)d_out);
}